// ADCGNN_amazon_81398220194637
// MI455X (gfx1250) — compile-verified
//
#include <hip/hip_runtime.h>
#include <hip/hip_bf16.h>
#include <math.h>

typedef float v2f __attribute__((ext_vector_type(2)));
typedef float v8f __attribute__((ext_vector_type(8)));

#define IN_DIM 128
#define HDIM 64

// ---------------------------------------------------------------------------
// Simple fill
// ---------------------------------------------------------------------------
__global__ __launch_bounds__(256) void fill_kernel(float* __restrict__ p, float v, int total) {
    int i = blockIdx.x * blockDim.x + threadIdx.x;
    if (i < total) p[i] = v;
}

// ---------------------------------------------------------------------------
// Degree: atomic count of dst occurrences
// ---------------------------------------------------------------------------
__global__ __launch_bounds__(256) void deg_kernel(const int* __restrict__ dst,
                                                  float* __restrict__ deg, int E) {
    int i = blockIdx.x * blockDim.x + threadIdx.x;
    if (i < E) atomicAdd(&deg[dst[i]], 1.0f);
}

__global__ __launch_bounds__(256) void dinv_kernel(const float* __restrict__ deg,
                                                   float* __restrict__ dinv, int n) {
    int i = blockIdx.x * blockDim.x + threadIdx.x;
    if (i < n) {
        float d = fmaxf(deg[i], 1.0f);
        dinv[i] = 1.0f / sqrtf(d);
    }
}

// ---------------------------------------------------------------------------
// WMMA f32 GEMM: out[n_rows x 64] = act(A[n_rows x K] @ W[K x 64] + bias)
// One wave per 16-row slab; 4 column tiles of 16; K-loop step 4 via
// V_WMMA_F32_16X16X4_F32. Requires n_rows % 16 == 0 (N=50000 = 3125*16).
// ---------------------------------------------------------------------------
__global__ __launch_bounds__(256) void gemm64_wmma_kernel(
    const float* __restrict__ A, const float* __restrict__ W,
    const float* __restrict__ bias, float* __restrict__ out,
    int n_rows, int K, int do_relu)
{
    const int lane  = threadIdx.x & 31;
    const int wave  = threadIdx.x >> 5;
    const int mbase = (blockIdx.x * 8 + wave) * 16;
    if (mbase >= n_rows) return;              // wave-uniform exit; EXEC stays all-1s

    const int half = lane >> 4;               // 0: lanes 0-15, 1: lanes 16-31
    const int l15  = lane & 15;

    v8f c[4];
    c[0] = {}; c[1] = {}; c[2] = {}; c[3] = {};

    const float* Arow = A + (size_t)(mbase + l15) * K;

    for (int k0 = 0; k0 < K; k0 += 4) {
        // A fragment (16x4): this lane holds A[m][k0+2*half .. k0+2*half+1]
        v2f a = *(const v2f*)(Arow + k0 + 2 * half);
#pragma unroll
        for (int nt = 0; nt < 4; ++nt) {
            // B fragment (4x16): comp0 = W[k0+2*half][col], comp1 = W[k0+2*half+1][col]
            const float* Wp = W + (size_t)(k0 + 2 * half) * HDIM + nt * 16 + l15;
            v2f b;
            b.x = Wp[0];
            b.y = Wp[HDIM];
            c[nt] = __builtin_amdgcn_wmma_f32_16x16x4_f32(
                false, a, false, b, (short)0, c[nt], false, false);
        }
    }

#pragma unroll
    for (int r = 0; r < 8; ++r) {
        const int m = mbase + r + 8 * half;
        float* orow = out + (size_t)m * HDIM;
#pragma unroll
        for (int nt = 0; nt < 4; ++nt) {
            const int col = nt * 16 + l15;
            float v = c[nt][r] + bias[col];
            if (do_relu) v = fmaxf(v, 0.0f);
            orow[col] = v;
        }
    }
}

// ---------------------------------------------------------------------------
// xs = x * dinv[row]   (elementwise over N*64)
// ---------------------------------------------------------------------------
__global__ __launch_bounds__(256) void scale_kernel(const float* __restrict__ x,
                                                    const float* __restrict__ dinv,
                                                    float* __restrict__ xs, int total) {
    int i = blockIdx.x * blockDim.x + threadIdx.x;
    if (i < total) xs[i] = x[i] * dinv[i >> 6];
}

// ---------------------------------------------------------------------------
// SpMM scatter: acc[dst[e]][:] += xs[src[e]][:]. One wave per edge,
// each lane owns a float2 (2*32 = 64 columns). Edge index loads are
// wave-uniform (scalarized by the compiler).
// ---------------------------------------------------------------------------
__global__ __launch_bounds__(256) void spmm_edge_kernel(const int* __restrict__ src,
                                                        const int* __restrict__ dst,
                                                        const float* __restrict__ xs,
                                                        float* __restrict__ acc, int E) {
    const int e    = (blockIdx.x * blockDim.x + threadIdx.x) >> 5;
    const int lane = threadIdx.x & 31;
    if (e >= E) return;
    const int s = src[e];
    const int d = dst[e];
    v2f v = *(const v2f*)(xs + (size_t)s * HDIM + lane * 2);
    float* ap = acc + (size_t)d * HDIM + lane * 2;
    atomicAdd(ap + 0, v.x);
    atomicAdd(ap + 1, v.y);
}

// ---------------------------------------------------------------------------
// out = x - acc * dinv[row]   (laplacian combine)
// ---------------------------------------------------------------------------
__global__ __launch_bounds__(256) void combine_kernel(const float* __restrict__ x,
                                                      const float* __restrict__ acc,
                                                      const float* __restrict__ dinv,
                                                      float* __restrict__ out, int total) {
    int i = blockIdx.x * blockDim.x + threadIdx.x;
    if (i < total) out[i] = x[i] - acc[i] * dinv[i >> 6];
}

// ---------------------------------------------------------------------------
// Per-node branch combination + attention softmax (over 3 branches) + mean.
// Wave per node, float2 per lane. Writes fin[node] = [attn_fused | mean_fused]
// as an N x 128 row (ready to be the A-matrix of the Wf1 WMMA GEMM).
// ---------------------------------------------------------------------------
__global__ __launch_bounds__(256) void attn_kernel(const float* __restrict__ f0,
                                                   const float* __restrict__ f1,
                                                   const float* __restrict__ f2,
                                                   const float* __restrict__ Wattn,
                                                   const float* __restrict__ battn,
                                                   float* __restrict__ fin, int n) {
    const int node = (blockIdx.x * blockDim.x + threadIdx.x) >> 5;
    const int lane = threadIdx.x & 31;
    if (node >= n) return;
    const size_t off = (size_t)node * HDIM + lane * 2;

    v2f x0 = *(const v2f*)(f0 + off);
    v2f x1 = *(const v2f*)(f1 + off);
    v2f x2 = *(const v2f*)(f2 + off);

    // THETAS = [[3,-3,0.75],[0,3,-1.5],[0,0,0.75]]
    v2f br0 = 3.0f * x0 - 3.0f * x1 + 0.75f * x2;
    v2f br1 = 3.0f * x1 - 1.5f * x2;
    v2f br2 = 0.75f * x2;

    v2f wa = *(const v2f*)(Wattn + lane * 2);
    float s0 = br0.x * wa.x + br0.y * wa.y;
    float s1 = br1.x * wa.x + br1.y * wa.y;
    float s2 = br2.x * wa.x + br2.y * wa.y;
#pragma unroll
    for (int o = 16; o >= 1; o >>= 1) {
        s0 += __shfl_xor(s0, o, 32);
        s1 += __shfl_xor(s1, o, 32);
        s2 += __shfl_xor(s2, o, 32);
    }
    const float bb = battn[0];
    s0 += bb; s1 += bb; s2 += bb;

    float mx = fmaxf(s0, fmaxf(s1, s2));
    float e0 = expf(s0 - mx), e1 = expf(s1 - mx), e2 = expf(s2 - mx);
    float inv = 1.0f / (e0 + e1 + e2);
    float w0 = e0 * inv, w1 = e1 * inv, w2 = e2 * inv;

    v2f attn = w0 * br0 + w1 * br1 + w2 * br2;
    v2f mean = (br0 + br1 + br2) * (1.0f / 3.0f);

    float* row = fin + (size_t)node * (2 * HDIM);
    *(v2f*)(row + lane * 2)        = attn;
    *(v2f*)(row + HDIM + lane * 2) = mean;
}

// ---------------------------------------------------------------------------
// fusion_weight = sigmoid(hidden @ Wf2 + bf2);
// fused = 0.1*fw*attn + (1-fw)*mean + 0.8*res. Wave per node.
// ---------------------------------------------------------------------------
__global__ __launch_bounds__(256) void fuse_kernel(const float* __restrict__ hidden,
                                                   const float* __restrict__ Wf2,
                                                   const float* __restrict__ bf2,
                                                   const float* __restrict__ fin,
                                                   const float* __restrict__ res,
                                                   float* __restrict__ fused, int n) {
    const int node = (blockIdx.x * blockDim.x + threadIdx.x) >> 5;
    const int lane = threadIdx.x & 31;
    if (node >= n) return;

    v2f h = *(const v2f*)(hidden + (size_t)node * HDIM + lane * 2);
    v2f w = *(const v2f*)(Wf2 + lane * 2);
    float p = h.x * w.x + h.y * w.y;
#pragma unroll
    for (int o = 16; o >= 1; o >>= 1) p += __shfl_xor(p, o, 32);
    p += bf2[0];
    float fw = 1.0f / (1.0f + expf(-p));

    const float* row = fin + (size_t)node * (2 * HDIM);
    v2f attn = *(const v2f*)(row + lane * 2);
    v2f mean = *(const v2f*)(row + HDIM + lane * 2);
    v2f r    = *(const v2f*)(res + (size_t)node * HDIM + lane * 2);

    v2f out = (0.1f * fw) * attn + (1.0f - fw) * mean + 0.8f * r;
    *(v2f*)(fused + (size_t)node * HDIM + lane * 2) = out;
}

// ---------------------------------------------------------------------------
// logits = h3 @ W4 + b4   (64 -> 2). Wave per node; W4 row-major 64x2 so
// rows 2l,2l+1 (4 floats) are contiguous -> one float4 per lane.
// ---------------------------------------------------------------------------
__global__ __launch_bounds__(256) void logits_kernel(const float* __restrict__ h3,
                                                     const float* __restrict__ W4,
                                                     const float* __restrict__ b4,
                                                     float* __restrict__ out, int n) {
    const int node = (blockIdx.x * blockDim.x + threadIdx.x) >> 5;
    const int lane = threadIdx.x & 31;
    if (node >= n) return;

    v2f h = *(const v2f*)(h3 + (size_t)node * HDIM + lane * 2);
    const float4 w = *(const float4*)(W4 + lane * 4);   // (2l,0)(2l,1)(2l+1,0)(2l+1,1)
    float p0 = h.x * w.x + h.y * w.z;
    float p1 = h.x * w.y + h.y * w.w;
#pragma unroll
    for (int o = 16; o >= 1; o >>= 1) {
        p0 += __shfl_xor(p0, o, 32);
        p1 += __shfl_xor(p1, o, 32);
    }
    if (lane == 0) {
        out[(size_t)node * 2 + 0] = p0 + b4[0];
        out[(size_t)node * 2 + 1] = p1 + b4[1];
    }
}

// ---------------------------------------------------------------------------
extern "C" void kernel_launch(void* const* d_in, const int* in_sizes, int n_in,
                              void* d_out, int out_size, void* d_ws, size_t ws_size,
                              hipStream_t stream) {
    const float* in_feat = (const float*)d_in[0];
    const int*   src     = (const int*)d_in[1];
    const int*   dst     = (const int*)d_in[2];
    const float* W1   = (const float*)d_in[3];
    const float* b1   = (const float*)d_in[4];
    const float* W2   = (const float*)d_in[5];
    const float* b2   = (const float*)d_in[6];
    const float* Wres = (const float*)d_in[7];
    const float* bres = (const float*)d_in[8];
    const float* Wattn = (const float*)d_in[9];
    const float* battn = (const float*)d_in[10];
    const float* Wf1  = (const float*)d_in[11];
    const float* bf1  = (const float*)d_in[12];
    const float* Wf2  = (const float*)d_in[13];
    const float* bf2  = (const float*)d_in[14];
    const float* W3   = (const float*)d_in[15];
    const float* b3   = (const float*)d_in[16];
    const float* W4   = (const float*)d_in[17];
    const float* b4   = (const float*)d_in[18];

    const int N = in_sizes[0] / IN_DIM;
    const int E = in_sizes[1];
    const int total64 = N * HDIM;

    // Workspace bump allocation (floats)
    float* ws = (float*)d_ws;
    size_t o = 0;
    float* deg    = ws + o; o += (size_t)N;
    float* dinv   = ws + o; o += (size_t)N;
    float* bufA   = ws + o; o += (size_t)N * HDIM;     // h_pre0 / xs / hidden / h3
    float* feat0  = ws + o; o += (size_t)N * HDIM;     // h_pre
    float* feat1  = ws + o; o += (size_t)N * HDIM;
    float* feat2  = ws + o; o += (size_t)N * HDIM;
    float* resb   = ws + o; o += (size_t)N * HDIM;
    float* accb   = ws + o; o += (size_t)N * HDIM;
    float* fin    = ws + o; o += (size_t)N * 2 * HDIM; // [attn | mean]
    float* fusedb = ws + o; o += (size_t)N * HDIM;

    const int T = 256;
    const int blkN    = (N + T - 1) / T;
    const int blkE    = (E + T - 1) / T;
    const int blk64   = (total64 + T - 1) / T;
    const int blkGemm = (N + 16 * 8 - 1) / (16 * 8);     // 8 waves/block, 16 rows/wave
    const int blkNode = (N + 8 - 1) / 8;                 // 8 wave-nodes per block
    const int blkEdge = (E + 8 - 1) / 8;                 // 8 wave-edges per block

    // degree -> d^-1/2
    fill_kernel<<<blkN, T, 0, stream>>>(deg, 0.0f, N);
    deg_kernel<<<blkE, T, 0, stream>>>(dst, deg, E);
    dinv_kernel<<<blkN, T, 0, stream>>>(deg, dinv, N);

    // h_pre = relu(relu(in_feat@W1+b1)@W2+b2); res = h_pre@Wres+bres
    gemm64_wmma_kernel<<<blkGemm, T, 0, stream>>>(in_feat, W1, b1, bufA, N, IN_DIM, 1);
    gemm64_wmma_kernel<<<blkGemm, T, 0, stream>>>(bufA, W2, b2, feat0, N, HDIM, 1);
    gemm64_wmma_kernel<<<blkGemm, T, 0, stream>>>(feat0, Wres, bres, resb, N, HDIM, 0);

    // feats[1] = laplacian(feats[0])
    scale_kernel<<<blk64, T, 0, stream>>>(feat0, dinv, bufA, total64);
    fill_kernel<<<blk64, T, 0, stream>>>(accb, 0.0f, total64);
    spmm_edge_kernel<<<blkEdge, T, 0, stream>>>(src, dst, bufA, accb, E);
    combine_kernel<<<blk64, T, 0, stream>>>(feat0, accb, dinv, feat1, total64);

    // feats[2] = laplacian(feats[1])
    scale_kernel<<<blk64, T, 0, stream>>>(feat1, dinv, bufA, total64);
    fill_kernel<<<blk64, T, 0, stream>>>(accb, 0.0f, total64);
    spmm_edge_kernel<<<blkEdge, T, 0, stream>>>(src, dst, bufA, accb, E);
    combine_kernel<<<blk64, T, 0, stream>>>(feat1, accb, dinv, feat2, total64);

    // branches + attention softmax + mean -> fin (N x 128)
    attn_kernel<<<blkNode, T, 0, stream>>>(feat0, feat1, feat2, Wattn, battn, fin, N);

    // hidden = relu(fin @ Wf1 + bf1)
    gemm64_wmma_kernel<<<blkGemm, T, 0, stream>>>(fin, Wf1, bf1, bufA, N, 2 * HDIM, 1);

    // gate + fuse (+0.8*res)
    fuse_kernel<<<blkNode, T, 0, stream>>>(bufA, Wf2, bf2, fin, resb, fusedb, N);

    // h3 = relu(fused @ W3 + b3)
    gemm64_wmma_kernel<<<blkGemm, T, 0, stream>>>(fusedb, W3, b3, bufA, N, HDIM, 1);

    // logits = h3 @ W4 + b4
    logits_kernel<<<blkNode, T, 0, stream>>>(bufA, W4, b4, (float*)d_out, N);
}